// LIF_R_42588895707321
// MI455X (gfx1250) — compile-verified
//
#include <hip/hip_runtime.h>
#include <cstdint>

#define N_DIM 8192

typedef __attribute__((ext_vector_type(2))) float v2f;
typedef __attribute__((ext_vector_type(4))) float v4f;
typedef __attribute__((ext_vector_type(8))) float v8f;

// The async-LDS builtin takes GCC-style int __vector(4) pointers,
// AS(1) for global and AS(3) for LDS, non-const (per hipcc diagnostics).
typedef int v4i_mem __attribute__((vector_size(16)));
typedef __attribute__((address_space(1))) v4i_mem* gas_ptr;
typedef __attribute__((address_space(3))) v4i_mem* lds_ptr;

#define HAS_WMMA_F32   __has_builtin(__builtin_amdgcn_wmma_f32_16x16x4_f32)
#define HAS_ASYNC_LDS  __has_builtin(__builtin_amdgcn_global_load_async_to_lds_b128)
#define HAS_WAIT_ASYNC __has_builtin(__builtin_amdgcn_s_wait_asynccnt)

__device__ inline gas_ptr to_global_v4i(const void* p) { return (gas_ptr)(uintptr_t)p; }
__device__ inline lds_ptr to_lds_v4i(void* p)          { return (lds_ptr)(uintptr_t)p; }

#if HAS_WMMA_F32
// One 8-wide K chunk: two V_WMMA_F32_16X16X4_F32 accumulations.
// Effective K assignment per WMMA is a permutation of the chunk; A and B use
// the same (half,vgpr)->k map, so the dot product is exact.
__device__ inline v8f wmma_step(v8f acc, v4f wa, v4f gb) {
    v2f a0 = {wa.x, wa.y}, b0 = {gb.x, gb.y};
    acc = __builtin_amdgcn_wmma_f32_16x16x4_f32(false, a0, false, b0,
                                                (short)0, acc, false, false);
    v2f a1 = {wa.z, wa.w}, b1 = {gb.z, gb.w};
    acc = __builtin_amdgcn_wmma_f32_16x16x4_f32(false, a1, false, b1,
                                                (short)0, acc, false, false);
    return acc;
}
#endif

// ---------------------------------------------------------------------------
// Kernel 1: partial matvec. Grid = 128*segCount blocks x 128 threads.
//   seg  = blockIdx.x >> 7  selects a K segment of length segLen,
//   each of the 4 waves in a block handles a 16-row strip over that segment.
// Partial sums go to I_part[seg*N + row]; kernel 2 reduces across segments.
// ---------------------------------------------------------------------------
__global__ __launch_bounds__(128) void lif_matvec_wmma(
    const float* __restrict__ w, const float* __restrict__ g,
    float* __restrict__ I_part, int segLen) {
    extern __shared__ float g_lds[];        // segLen floats (<= 32 KB)

    const int tid  = threadIdx.x;
    const int lane = tid & 31;
    const int wid  = tid >> 5;              // 4 waves per block
    const int seg  = blockIdx.x >> 7;
    const int blk  = blockIdx.x & 127;
    const int kBase = seg * segLen;

    // ---- Stage this K segment of g into LDS (async DMA path) ----
#if HAS_ASYNC_LDS
    for (int i = tid * 4; i < segLen; i += 128 * 4) {
        __builtin_amdgcn_global_load_async_to_lds_b128(
            to_global_v4i(g + kBase + i), to_lds_v4i(&g_lds[i]), 0, 0);
    }
#if HAS_WAIT_ASYNC
    __builtin_amdgcn_s_wait_asynccnt(0);
#endif
#else
    for (int i = tid * 4; i < segLen; i += 128 * 4) {
        *(v4f*)&g_lds[i] = *(const v4f*)&g[kBase + i];
    }
#endif
    __syncthreads();

    // ---- Per-wave 16-row strip ----
    const int rowBase = blk * 64 + wid * 16;
    const int m    = lane & 15;             // row within strip
    const int half = lane >> 4;             // A layout: lanes 0-15 K0/K1, 16-31 K2/K3
    const int goff = 4 * half;
    const float* wrow = w + (size_t)(rowBase + m) * N_DIM + kBase + goff;

#if HAS_WMMA_F32
    v8f acc = {0.f, 0.f, 0.f, 0.f, 0.f, 0.f, 0.f, 0.f};
    // 32 K-values (=128 B per row, one cache line) per iteration:
    // batch 4 global b128 + 4 ds b128 loads, then 8 WMMAs.
    for (int k0 = 0; k0 < segLen; k0 += 32) {
        const v4f wa0 = *(const v4f*)(wrow + k0);
        const v4f wa1 = *(const v4f*)(wrow + k0 + 8);
        const v4f wa2 = *(const v4f*)(wrow + k0 + 16);
        const v4f wa3 = *(const v4f*)(wrow + k0 + 24);
        const v4f gb0 = *(const v4f*)&g_lds[k0 + goff];
        const v4f gb1 = *(const v4f*)&g_lds[k0 + 8 + goff];
        const v4f gb2 = *(const v4f*)&g_lds[k0 + 16 + goff];
        const v4f gb3 = *(const v4f*)&g_lds[k0 + 24 + goff];
        __builtin_prefetch(wrow + k0 + 1024, 0, 3);   // ~4KB ahead, keep near

        acc = wmma_step(acc, wa0, gb0);
        acc = wmma_step(acc, wa1, gb1);
        acc = wmma_step(acc, wa2, gb2);
        acc = wmma_step(acc, wa3, gb3);
    }
    // C/D layout: VGPR i holds row M=i at lanes 0-15 and row M=i+8 at lanes
    // 16-31; all N columns equal (B columns identical) -> read N=0 column.
    float* outp = I_part + (size_t)seg * N_DIM + rowBase;
    if (lane == 0) {
#pragma unroll
        for (int i = 0; i < 8; ++i) outp[i] = acc[i];
    } else if (lane == 16) {
#pragma unroll
        for (int i = 0; i < 8; ++i) outp[8 + i] = acc[i];
    }
#else
    float a = 0.f;
    for (int k0 = 0; k0 < segLen; k0 += 8) {
        const v4f wa = *(const v4f*)(wrow + k0);
        const v4f gb = *(const v4f*)&g_lds[k0 + goff];
        a = fmaf(wa.x, gb.x, a); a = fmaf(wa.y, gb.y, a);
        a = fmaf(wa.z, gb.z, a); a = fmaf(wa.w, gb.w, a);
    }
    a += __shfl_xor(a, 16, 32);
    if (lane < 16) I_part[(size_t)seg * N_DIM + rowBase + lane] = a;
#endif
}

// ---------------------------------------------------------------------------
// Kernel 2: reduce K-segment partials + elementwise LIF update.
// Only (v_new, spiked_soft) are returned by the reference.
// ---------------------------------------------------------------------------
__global__ __launch_bounds__(256) void lif_update(
    const float* __restrict__ x_in, const float* __restrict__ v,
    const float* __restrict__ theta_s, const float* __restrict__ v_rest,
    const float* __restrict__ I_part, int segCount, float* __restrict__ out) {
    const int i = blockIdx.x * blockDim.x + threadIdx.x;
    if (i >= N_DIM) return;

    float I = x_in[i];
    for (int s = 0; s < segCount; ++s) I += I_part[(size_t)s * N_DIM + i];

    const float vi = v[i];
    const float vr = v_rest[i];
    const float th = theta_s[i];

    const float v_next = vi + (vr - vi + I * (1.0f / 0.05f));   // dv; C_M = 0.05
    const float soft   = 1.0f / (1.0f + expf(-(v_next - th)));  // sigmoid
    const bool  spiked = v_next >= th;
    const float v_rst  = vr + 0.15f * (vi - vr) - 12.0f;        // F_V, DELTA_V

    out[i]         = spiked ? v_rst : v_next;   // v_new
    out[N_DIM + i] = soft;                      // spiked_soft
}

// ---------------------------------------------------------------------------
extern "C" void kernel_launch(void* const* d_in, const int* in_sizes, int n_in,
                              void* d_out, int out_size, void* d_ws, size_t ws_size,
                              hipStream_t stream) {
    // setup_inputs order: x_in, v, g, theta_s, w, v_rest, tau_g  (all fp32)
    const float* x_in    = (const float*)d_in[0];
    const float* v       = (const float*)d_in[1];
    const float* g       = (const float*)d_in[2];
    const float* theta_s = (const float*)d_in[3];
    const float* w       = (const float*)d_in[4];
    const float* v_rest  = (const float*)d_in[5];
    (void)d_in[6]; // tau_g only feeds hidden state g_new, which is not returned

    float* out    = (float*)d_out;
    float* I_part = (float*)d_ws;

    // Split K into as many segments as the workspace allows (pow2, max 8):
    // more segments -> more waves in flight -> closer to the HBM roofline.
    const size_t per = (size_t)N_DIM * sizeof(float);
    int segCount = 1;
    while (segCount < 8 && (size_t)(segCount * 2) * per <= ws_size) segCount *= 2;
    const int segLen = N_DIM / segCount;

    lif_matvec_wmma<<<128 * segCount, 128, segLen * sizeof(float), stream>>>(
        w, g, I_part, segLen);
    lif_update<<<(N_DIM + 255) / 256, 256, 0, stream>>>(
        x_in, v, theta_s, v_rest, I_part, segCount, out);
}